// EulerMatrixGenerator_17918603559171
// MI455X (gfx1250) — compile-verified
//
#include <hip/hip_runtime.h>

// EulerMatrixGenerator for MI455X (gfx1250).
//
// The reference assembles a 12288x12288 fp32 matrix (604 MB) that is
// analytically sparse: all component matrices are (bi)diagonal, so every
// nonzero of A has a closed form. The workload is store-bandwidth bound
// (604 MB / 23.3 TB/s ~= 26 us floor); matrix cores are irrelevant here.
// We use the CDNA5 async LDS->global store engine (ASYNCcnt) for the bulk
// zero-fill (GVS addressing + non-temporal hint), then scatter the ~61K
// nonzeros with closed-form math.

#define NG   64                 // n
#define NN   (NG * NG)          // 4096
#define DIM  (3 * NN)           // 12288
#define LRF  0.001f

typedef __attribute__((ext_vector_type(4))) float v4f;

// ---------------------------------------------------------------------------
// Kernel 1: invdx = 1 / (DT/CN * max(c) * sqrt(2)) -> ws[0]
// ---------------------------------------------------------------------------
__global__ void euler_cmax(const float* __restrict__ c, float* __restrict__ ws) {
    __shared__ float red[256];
    int t = threadIdx.x;
    float m = -__builtin_huge_valf();
    for (int i = t; i < NN; i += 256) m = fmaxf(m, c[i]);
    red[t] = m;
    __syncthreads();
    for (int s = 128; s > 0; s >>= 1) {
        if (t < s) red[t] = fmaxf(red[t], red[t + s]);
        __syncthreads();
    }
    if (t == 0) {
        float dx = 10.0f * red[0] * 1.41421356237309515f;  // DT/CN = 10
        ws[0] = 1.0f / dx;
    }
}

// ---------------------------------------------------------------------------
// Kernel 2: zero-fill 604 MB of output through the CDNA5 async data path.
// A 4KB block of zeros lives in LDS; each lane issues
// global_store_async_from_lds_b128 (16B/lane) in a grid-stride loop using
// GVS addressing (SGPR64 base + 32-bit VGPR byte offset) so the inner loop
// is just a shift + async-store issue. th:TH_STORE_NT streams past L2
// (output is 3x L2 capacity and never re-read on device). ASYNCcnt caps
// in-flight ops automatically; drain before exit.
// Generic LDS pointer low 32 bits == WG-relative LDS offset (ISA 10.2).
// ---------------------------------------------------------------------------
__global__ void euler_zero_async(float* __restrict__ out, int n4) {
    __shared__ v4f zbuf[256];
    v4f z = {0.0f, 0.0f, 0.0f, 0.0f};
    zbuf[threadIdx.x] = z;
    __syncthreads();  // s_wait_dscnt 0 + barrier: LDS zeros visible

    unsigned ldsoff = (unsigned)(unsigned long long)&zbuf[threadIdx.x];
    int tid    = blockIdx.x * blockDim.x + threadIdx.x;
    int stride = gridDim.x * blockDim.x;
    unsigned long long base = (unsigned long long)out;

    for (int i = tid; i < n4; i += stride) {
        unsigned goff = (unsigned)i << 4;   // byte offset, < 604MB
        asm volatile("global_store_async_from_lds_b128 %0, %1, %2 th:TH_STORE_NT"
                     :: "v"(goff), "v"(ldsoff), "s"(base) : "memory");
    }
    asm volatile("s_wait_asynccnt 0" ::: "memory");
}

// ---------------------------------------------------------------------------
// Kernel 3: scatter the closed-form nonzeros. One thread per g in [0, 4096).
// Flattening: g = i*n + ii  ->  field.T.flat, i.e. cflat[g] = c[ii][i].
// ---------------------------------------------------------------------------
__global__ void euler_fill(const float* __restrict__ c,
                           const float* __restrict__ kp,
                           const float* __restrict__ kx,
                           const float* __restrict__ ky,
                           const float* __restrict__ dk,
                           const float* __restrict__ ws,
                           float* __restrict__ out) {
    int g = blockIdx.x * blockDim.x + threadIdx.x;
    if (g >= NN) return;

    const float invdx  = ws[0];
    const float invdx2 = invdx * invdx;

    int ii  = g & (NG - 1);      // row within block
    int i   = g >> 6;            // block index
    int idx = ii * NG + i;       // row-major (ii, i)

    float cg  = c[idx];
    float dkv = dk[idx];
    float kxf = kx[idx] - LRF * dkv;
    float kyf = ky[idx] - LRF * dkv;
    float kpv = kp[ii];

    float minv  = 1.0f / (1.0f - 0.5f * kpv);   // 1/Mdiag
    float Mp    = 1.0f + 0.5f * kpv;            // Mprime diag
    float nxinv = 1.0f / (1.0f + 0.5f * kxf);
    float Nxp   = 1.0f - 0.5f * kxf;
    float nyinv = 1.0f / (1.0f + 0.5f * kyf);
    float Nyp   = 1.0f - 0.5f * kyf;

    bool has_pn = (i  < NG - 1);   // g+n exists (block superdiag)
    bool has_mn = (i  > 0);        // g-n exists
    bool has_p1 = (ii < NG - 1);   // g+1 exists (within-block superdiag)
    bool has_m1 = (ii > 0);        // g-1 exists

    float c_pn = has_pn ? c[idx + 1]  : 0.0f;   // cflat[g+n] = c[ii][i+1]
    float c_mn = has_mn ? c[idx - 1]  : 0.0f;   // cflat[g-n] = c[ii][i-1]
    float c_p1 = has_p1 ? c[idx + NG] : 0.0f;   // cflat[g+1] = c[ii+1][i]

    float nxinv_pn = 0.0f, Nxp_pn = 0.0f;
    if (has_pn) {
        float kxn = kx[idx + 1] - LRF * dk[idx + 1];
        nxinv_pn  = 1.0f / (1.0f + 0.5f * kxn);
        Nxp_pn    = 1.0f - 0.5f * kxn;
    }
    float nyinv_p1 = 0.0f, Nyp_p1 = 0.0f;
    if (has_p1) {
        float kyn = ky[idx + NG] - LRF * dk[idx + NG];
        nyinv_p1  = 1.0f / (1.0f + 0.5f * kyn);
        Nyp_p1    = 1.0f - 0.5f * kyn;
    }

    long long r1 = (long long)g * DIM;               // A1x rows
    long long r2 = (long long)(NN + g) * DIM;        // A2x rows
    long long r3 = (long long)(2 * NN + g) * DIM;    // A3x rows

    float cg2 = cg * cg;

    // (Tx@Sx)[g,g] and (Ty@Sy)[g,g]
    float txsx_d = -cg2 * invdx2 * nxinv
                   - (has_pn ? cg2 * invdx2 * nxinv_pn : 0.0f);
    float tysy_d = -cg2 * invdx2 * nyinv
                   - (has_p1 ? c_p1 * c_p1 * invdx2 * nyinv_p1 : 0.0f);

    // ---- A11 ----
    out[r1 + g] = minv * Mp + minv * (txsx_d + tysy_d);
    if (has_p1) out[r1 + g + 1]  = minv * (c_p1 * c_p1 * invdx2 * nyinv_p1);
    if (has_m1) out[r1 + g - 1]  = minv * (cg2 * invdx2 * nyinv);
    if (has_pn) out[r1 + g + NG] = minv * (cg * c_pn * invdx2 * nxinv_pn);
    if (has_mn) out[r1 + g - NG] = minv * (cg * c_mn * invdx2 * nxinv);

    // ---- A12 = minv * Rx * (nxinv*Nxp) ----
    out[r1 + NN + g] = minv * cg * invdx * nxinv * Nxp;
    if (has_pn) out[r1 + NN + g + NG] = -minv * cg * invdx * nxinv_pn * Nxp_pn;

    // ---- A13 = minv * Ry * (nyinv*Nyp) ----
    out[r1 + 2 * NN + g] = minv * cg * invdx * nyinv * Nyp;
    if (has_p1) out[r1 + 2 * NN + g + 1] = -minv * c_p1 * invdx * nyinv_p1 * Nyp_p1;

    // ---- A21 = nxinv * Sx ----
    out[r2 + g] = -nxinv * cg * invdx;
    if (has_mn) out[r2 + g - NG] = nxinv * c_mn * invdx;

    // ---- A22 = diag(nxinv * Nxp) ----
    out[r2 + NN + g] = nxinv * Nxp;

    // ---- A31 = nyinv * Sy ----
    out[r3 + g] = -nyinv * cg * invdx;
    if (has_m1) out[r3 + g - 1] = nyinv * cg * invdx;   // Sy subdiag uses current row's c

    // ---- A33 = diag(nyinv * Nyp) ----
    out[r3 + 2 * NN + g] = nyinv * Nyp;
}

// ---------------------------------------------------------------------------
extern "C" void kernel_launch(void* const* d_in, const int* in_sizes, int n_in,
                              void* d_out, int out_size, void* d_ws, size_t ws_size,
                              hipStream_t stream) {
    const float* c  = (const float*)d_in[0];
    const float* kp = (const float*)d_in[1];
    const float* kx = (const float*)d_in[2];
    const float* ky = (const float*)d_in[3];
    const float* dk = (const float*)d_in[4];
    float* out = (float*)d_out;
    float* ws  = (float*)d_ws;

    // 1) scalar reduction -> ws[0] = 1/dx
    euler_cmax<<<1, 256, 0, stream>>>(c, ws);

    // 2) async-engine zero-fill of all 12288*12288 floats (B128 granularity)
    int n4 = out_size / 4;   // 37,748,736 float4 chunks
    euler_zero_async<<<8192, 256, 0, stream>>>(out, n4);

    // 3) scatter closed-form nonzeros (depends on ws[0] and on zero-fill)
    euler_fill<<<(NN + 255) / 256, 256, 0, stream>>>(c, kp, kx, ky, dk, ws, out);
}